// MaxMinQuantizer_43628277793222
// MI455X (gfx1250) — compile-verified
//
#include <hip/hip_runtime.h>
#include <stdint.h>

// ---------------------------------------------------------------------------
// NVFP4 (e2m1 data + e4m3 block scale) quantize->dequantize, fp32 in/out.
// HBM-bound streamer: 512 MiB traffic @ 23.3 TB/s => ~22us floor.
// gfx1250 async global->LDS staging (ASYNCcnt) with double buffering.
// ---------------------------------------------------------------------------

#define TPB 256
#define BLOCKS_PER_TILE 256                        // fp4 blocks (16 floats) per tile
#define TILE_FLOATS (BLOCKS_PER_TILE * 16)         // 4096 floats
#define TILE_BYTES  (TILE_FLOATS * 4)              // 16384 bytes
#define LDS_BLK_STRIDE 80                          // 64B block + 16B pad (bank spread)
#define TILE_LDS (BLOCKS_PER_TILE * LDS_BLK_STRIDE)   // 20480 bytes per buffer
#define TILE_V4  (TILE_LDS / 16)                   // 1280 v4f per buffer

typedef float v4f __attribute__((ext_vector_type(4)));

// One 16-byte async chunk. lds_addr MUST be derived from the shared array's
// address so the array provably escapes (otherwise LLVM proves the LDS global
// is never written and folds the subsequent ds_loads to undef).
__device__ __forceinline__ void async_chunk16(uint32_t lds_addr, uint32_t voff,
                                              unsigned long long sbase) {
  asm volatile("global_load_async_to_lds_b128 %0, %1, %2"
               :: "v"(lds_addr), "v"(voff), "s"(sbase)
               : "memory");
}

// Round-to-nearest-even cast of a (>=0) to the e4m3 magnitude grid, clamp 448.
__device__ __forceinline__ float e4m3_q(float a) {
  a = fminf(a, 448.0f);
  if (a < 0.015625f) {                        // below 2^-6: e4m3 subnormal, quantum 2^-9
    return rintf(a * 512.0f) * 0.001953125f;
  }
  uint32_t u = __float_as_uint(a);
  u += 0x7FFFFu + ((u >> 20) & 1u);           // RNE to 3 mantissa bits
  u &= 0xFFF00000u;
  return __uint_as_float(u);
}

// Round x/scale to the signed e2m1 grid and dequantize, division-free:
// compare |x| against mids*scale (side='right': ties go up), pick grid*scale.
__device__ __forceinline__ float fp4_qdq1(float x, float s) {
  float ax = fabsf(x);
  float r =
      (ax < 2.5f * s)
          ? ((ax < 1.25f * s)
                 ? ((ax < 0.75f * s) ? ((ax < 0.25f * s) ? 0.0f : 0.5f * s) : s)
                 : ((ax < 1.75f * s) ? 1.5f * s : 2.0f * s))
          : ((ax < 3.5f * s) ? 3.0f * s
                             : ((ax < 5.0f * s) ? 4.0f * s : 6.0f * s));
  return copysignf(r, x);
}

// Quantize-dequantize one 16-float block given its 4 v4f values; NT stores.
__device__ __forceinline__ void qdq_block16v(v4f a, v4f b, v4f c, v4f d,
                                             v4f* out4, float inv6sf) {
  float m0 = fmaxf(fabsf(a.x), fabsf(a.y));
  float m1 = fmaxf(fabsf(a.z), fabsf(a.w));
  float m2 = fmaxf(fabsf(b.x), fabsf(b.y));
  float m3 = fmaxf(fabsf(b.z), fabsf(b.w));
  float m4 = fmaxf(fabsf(c.x), fabsf(c.y));
  float m5 = fmaxf(fabsf(c.z), fabsf(c.w));
  float m6 = fmaxf(fabsf(d.x), fabsf(d.y));
  float m7 = fmaxf(fabsf(d.z), fabsf(d.w));
  float m = fmaxf(fmaxf(fmaxf(m0, m1), fmaxf(m2, m3)),
                  fmaxf(fmaxf(m4, m5), fmaxf(m6, m7)));
  float scale = e4m3_q(fabsf(m * inv6sf));
  v4f oa, ob, oc, od;
  oa.x = fp4_qdq1(a.x, scale); oa.y = fp4_qdq1(a.y, scale);
  oa.z = fp4_qdq1(a.z, scale); oa.w = fp4_qdq1(a.w, scale);
  ob.x = fp4_qdq1(b.x, scale); ob.y = fp4_qdq1(b.y, scale);
  ob.z = fp4_qdq1(b.z, scale); ob.w = fp4_qdq1(b.w, scale);
  oc.x = fp4_qdq1(c.x, scale); oc.y = fp4_qdq1(c.y, scale);
  oc.z = fp4_qdq1(c.z, scale); oc.w = fp4_qdq1(c.w, scale);
  od.x = fp4_qdq1(d.x, scale); od.y = fp4_qdq1(d.y, scale);
  od.z = fp4_qdq1(d.z, scale); od.w = fp4_qdq1(d.w, scale);
  __builtin_nontemporal_store(oa, out4 + 0);   // 512MB stream > 192MB L2: bypass
  __builtin_nontemporal_store(ob, out4 + 1);
  __builtin_nontemporal_store(oc, out4 + 2);
  __builtin_nontemporal_store(od, out4 + 3);
}

__global__ __launch_bounds__(TPB) void nvfp4_qdq_kernel(
    const float* __restrict__ x, const float* __restrict__ sfp,
    float* __restrict__ out, long long n, int ntiles) {
  __shared__ __align__(16) v4f smem4[2 * TILE_V4];   // 40KB -> 8 WGs/WGP by LDS

  const int tid = (int)threadIdx.x;
  const float inv6sf = sfp[0] * (1.0f / 6.0f);

  // LDS byte address of the shared buffer start (flat addr low 32 = LDS offset).
  // Deriving asm operands from this makes smem4 escape -> ds reads survive.
  const uint32_t lds_base = (uint32_t)(uintptr_t)(void*)&smem4[0];

  // Staging map: thread tid, chunk j covers tile granule q = j*256 + tid (16B units).
  // LDS dest = (q>>2)*80 + (q&3)*16  => per-j increment 5120; voffset increment 4096.
  const uint32_t lds_st =
      lds_base + (uint32_t)(tid >> 2) * LDS_BLK_STRIDE + (uint32_t)(tid & 3) * 16u;
  const uint32_t v_st = (uint32_t)tid * 16u;
  const int tstride = (int)gridDim.x;

  int tile = (int)blockIdx.x;
  if (tile < ntiles) {  // prologue: stage first tile into buffer 0
    unsigned long long base =
        (unsigned long long)(uintptr_t)x + (unsigned long long)tile * TILE_BYTES;
    uint32_t l = lds_st, v = v_st;
#pragma unroll
    for (int j = 0; j < 4; ++j) { async_chunk16(l, v, base); l += 5120u; v += 4096u; }
  }

  uint32_t buf = 0;
  for (; tile < ntiles; tile += tstride) {
    const int next = tile + tstride;
    if (next < ntiles) {
      // Safe to stage into buf^1: the trailing barrier of the tile that last
      // used it was already passed by every wave in this workgroup.
      unsigned long long nb =
          (unsigned long long)(uintptr_t)x + (unsigned long long)next * TILE_BYTES;
      uint32_t l = (buf ^ 1u) * (uint32_t)TILE_LDS + lds_st;
      uint32_t v = v_st;
#pragma unroll
      for (int j = 0; j < 4; ++j) { async_chunk16(l, v, nb); l += 5120u; v += 4096u; }
      // async loads complete in order: <=4 outstanding => current tile landed
      asm volatile("s_wait_asynccnt 4" ::: "memory");
    } else {
      asm volatile("s_wait_asynccnt 0" ::: "memory");
    }
    __syncthreads();  // all waves' staging of current tile visible

    // Lane processes its own 16-float block: 64B at padded stride 5 v4f (80B).
    const v4f* lin = &smem4[buf * (uint32_t)TILE_V4 + (uint32_t)tid * 5u];
    v4f a = lin[0], b = lin[1], c = lin[2], d = lin[3];
    v4f* oto = (v4f*)(out + (size_t)tile * TILE_FLOATS) + (size_t)tid * 4;
    qdq_block16v(a, b, c, d, oto, inv6sf);

    __syncthreads();  // everyone done reading this buffer before it is restaged
    buf ^= 1u;
  }

  // Remainder blocks (n not a multiple of TILE_FLOATS); n % 16 == 0 per reference.
  long long blk = (long long)ntiles * BLOCKS_PER_TILE +
                  (long long)blockIdx.x * TPB + tid;
  const long long nblk = n >> 4;
  const long long bstride = (long long)tstride * TPB;
  for (; blk < nblk; blk += bstride) {
    const v4f* ip = (const v4f*)(x + blk * 16);
    v4f a = __builtin_nontemporal_load(ip + 0);
    v4f b = __builtin_nontemporal_load(ip + 1);
    v4f c = __builtin_nontemporal_load(ip + 2);
    v4f d = __builtin_nontemporal_load(ip + 3);
    v4f* op = (v4f*)(out + blk * 16);
    qdq_block16v(a, b, c, d, op, inv6sf);
  }
}

extern "C" void kernel_launch(void* const* d_in, const int* in_sizes, int n_in,
                              void* d_out, int out_size, void* d_ws, size_t ws_size,
                              hipStream_t stream) {
  (void)n_in; (void)out_size; (void)d_ws; (void)ws_size;
  const float* x  = (const float*)d_in[0];
  const float* sf = (const float*)d_in[1];
  float* out = (float*)d_out;
  long long n = (long long)in_sizes[0];
  int ntiles = (int)(n / TILE_FLOATS);              // 16384 for 8192x8192
  int grid = ntiles >= 2048 ? 2048 : (ntiles > 0 ? ntiles : 1);
  nvfp4_qdq_kernel<<<dim3(grid), dim3(TPB), 0, stream>>>(x, sf, out, n, ntiles);
}